// EmbracementLayer_38534446579794
// MI455X (gfx1250) — compile-verified
//
#include <hip/hip_runtime.h>
#include <hip/hip_bf16.h>

// out[b, e] = x[b, idx[b, e], e]
// x:   [BS, SEQ_LEN, EMB] f32
// idx: [BS, EMB]          i32
// out: [BS, EMB]          f32
#define BS      32
#define SEQ_LEN 4096
#define EMB     1024

// Native clang vector types: __builtin_nontemporal_load/store require
// ext_vector_type (HIP's int4/float4 structs are rejected).
typedef int   v4i __attribute__((ext_vector_type(4)));
typedef float v4f __attribute__((ext_vector_type(4)));

// Grid = BS blocks of 256 threads: block b handles all EMB=1024 outputs of
// batch row b (4 per thread). All row bases are wave-uniform -> SGPR base +
// 32-bit VGPR byte offset (GVS addressing), so per-gather address math is a
// single 32-bit VALU op instead of 64-bit shift+add pairs.
//
// Per thread: one coalesced b128 NT idx load, 4 independent scattered b32 NT
// gathers (MLP=4 behind one s_wait_loadcnt), one coalesced b128 NT store.
__global__ __launch_bounds__(256) void
embrace_gather_kernel(const float* __restrict__ x,
                      const int*   __restrict__ idx,
                      float*       __restrict__ out) {
    const int b  = blockIdx.x;            // batch row, uniform per block
    const int e0 = threadIdx.x << 2;      // first of 4 embedding columns

    // Wave-uniform row bases (SGPR-resident).
    const int*   __restrict__ idx_b = idx + b * EMB;
    float*       __restrict__ out_b = out + b * EMB;
    const char*  __restrict__ xb =
        (const char*)(x + (size_t)b * (size_t)SEQ_LEN * (size_t)EMB);

    // Coalesced vector load of 4 indices (global_load_b128, th:TH_LOAD_NT).
    const v4i s4 = __builtin_nontemporal_load(
        reinterpret_cast<const v4i*>(idx_b + e0));

    // 32-bit byte offsets within the 16 MB row slab:
    //   off = s*EMB*4 + e*4 = (s << 12) | (e << 2), max < 2^26.
    const unsigned o0 = ((unsigned)s4.x << 12) | (unsigned)((e0 + 0) << 2);
    const unsigned o1 = ((unsigned)s4.y << 12) | (unsigned)((e0 + 1) << 2);
    const unsigned o2 = ((unsigned)s4.z << 12) | (unsigned)((e0 + 2) << 2);
    const unsigned o3 = ((unsigned)s4.w << 12) | (unsigned)((e0 + 3) << 2);

    // 4 independent scattered gathers: saddr(SGPR pair) + 32-bit voffset.
    const float v0 = __builtin_nontemporal_load((const float*)(xb + o0));
    const float v1 = __builtin_nontemporal_load((const float*)(xb + o1));
    const float v2 = __builtin_nontemporal_load((const float*)(xb + o2));
    const float v3 = __builtin_nontemporal_load((const float*)(xb + o3));

    // Coalesced vector store (global_store_b128, th:TH_STORE_NT).
    v4f r;
    r.x = v0; r.y = v1; r.z = v2; r.w = v3;
    __builtin_nontemporal_store(r, reinterpret_cast<v4f*>(out_b + e0));
}

extern "C" void kernel_launch(void* const* d_in, const int* in_sizes, int n_in,
                              void* d_out, int out_size, void* d_ws, size_t ws_size,
                              hipStream_t stream) {
    (void)in_sizes; (void)n_in; (void)d_ws; (void)ws_size; (void)out_size;

    const float* x   = (const float*)d_in[0];   // [BS, SEQ_LEN, EMB] f32
    const int*   idx = (const int*)  d_in[1];   // [BS, EMB] i32
    float*       out = (float*)d_out;           // [BS, EMB] f32

    // One block per batch row; 256 threads (8 wave32s) x 4 outputs = EMB.
    embrace_gather_kernel<<<BS, 256, 0, stream>>>(x, idx, out);
}